// QLSTM_65481071400269
// MI455X (gfx1250) — compile-verified
//
#include <hip/hip_runtime.h>
#include <math.h>

#define SEQ 64
#define BATCH 128
#define IN_DIM 128
#define HID 256
#define NQ 8
#define DIM 256
#define COMB 384

typedef float v2f __attribute__((ext_vector_type(2)));
typedef float v8f __attribute__((ext_vector_type(8)));

struct QParams {
  const float* W[4];
  const float* b[4];
  const float* q[4];
  const float* U[4];
  const float* c[4];
};

// ---- workspace layout (float offsets) ----
// A4 : 4 gates * 512*512, packed per-wave WMMA A-operand tile layout
// X  : 4 gates * 512*128, packed per-wave WMMA B-operand layout (v2f pairs)
// EP : 4*32*8*128 per-(gate,mtile,qubit,sample) expectation partials
// H,C: 128*256 each
#define OFF_A4 0
#define OFF_X  1048576
#define OFF_EP 1310720
#define OFF_H  1441792
#define OFF_C  1474560

__global__ void qlstm_zero(float* p, int n) {
  int i = blockIdx.x * blockDim.x + threadIdx.x;
  if (i < n) p[i] = 0.f;
}

// ---------------------------------------------------------------------------
// Packed B-operand store helper: amplitude row pair (2a,2a+1)=(Re,Im) of
// column n lands in the v2f read by lane (a&1)*16+(n&15) at k-step a>>1.
// ---------------------------------------------------------------------------
__device__ __forceinline__ void store_psi(v2f* PB, int g, int nt, int lr,
                                          int a, float re, float im) {
  v2f w; w.x = re; w.y = im;
  PB[((size_t)(g * 8 + nt) * 128 + (a >> 1)) * 32 + ((a & 1) << 4) + lr] = w;
}

// ---------------------------------------------------------------------------
// k0: build the 4 fixed 256x256 complex unitaries by simulating the circuit
// on each basis column, then scatter into the packed real 512x512 A-matrix.
// A'[2a][2b]=Vr, A'[2a][2b+1]=-Vi, A'[2a+1][2b]=Vi, A'[2a+1][2b+1]=Vr.
// Packed A index: (((g*32+mt)*128+ks)*32+lane)*2+j with lane=(rem>>1)*16+(m&15),
// ks=k>>2, rem=k&3, j=rem&1  (ISA 16x4 f32 A-operand layout).
// ---------------------------------------------------------------------------
__global__ __launch_bounds__(64) void qlstm_build_unitary(QParams P, float* __restrict__ A4) {
  int gid = blockIdx.x * 64 + threadIdx.x;  // 0..1023
  int g   = gid >> 8;
  int col = gid & 255;
  const float* qp = P.q[g];  // (8, 2, 3)
  float sr[256], si[256];
  for (int a = 0; a < 256; ++a) { sr[a] = 0.f; si[a] = 0.f; }
  sr[col] = 1.f;
  for (int d = 0; d < 2; ++d) {
    for (int q = 0; q < NQ; ++q) {
      int m = 1 << (7 - q);  // qubit q acts on bit (7-q) of the flat index
      float tx = qp[(q * 2 + d) * 3 + 0];
      float ty = qp[(q * 2 + d) * 3 + 1];
      float tz = qp[(q * 2 + d) * 3 + 2];
      float cx_ = cosf(0.5f * tx), sx = sinf(0.5f * tx);
      float cy  = cosf(0.5f * ty), sy = sinf(0.5f * ty);
      float cz  = cosf(0.5f * tz), sz = sinf(0.5f * tz);
      for (int a0 = 0; a0 < 256; ++a0) {
        if (a0 & m) continue;
        int a1 = a0 | m;
        float r0 = sr[a0], i0 = si[a0], r1 = sr[a1], i1 = si[a1];
        // RX: new0 = c*v0 - i*s*v1 ; new1 = -i*s*v0 + c*v1
        float nr0 = cx_ * r0 + sx * i1, ni0 = cx_ * i0 - sx * r1;
        float nr1 = sx * i0 + cx_ * r1, ni1 = -sx * r0 + cx_ * i1;
        // RY
        r0 = cy * nr0 - sy * nr1; i0 = cy * ni0 - sy * ni1;
        r1 = sy * nr0 + cy * nr1; i1 = sy * ni0 + cy * ni1;
        // RZ: v0 *= (cz - i*sz) ; v1 *= (cz + i*sz)
        nr0 = cz * r0 + sz * i0;  ni0 = cz * i0 - sz * r0;
        nr1 = cz * r1 - sz * i1;  ni1 = cz * i1 + sz * r1;
        sr[a0] = nr0; si[a0] = ni0; sr[a1] = nr1; si[a1] = ni1;
      }
    }
    // CNOT ring: (q,q+1) for q<7, then (7,0); qubit q -> bit (7-q)
    for (int q = 0; q < NQ; ++q) {
      int bc = 7 - q;
      int bt = (q == 7) ? 7 : (6 - q);
      int mc = 1 << bc, mt = 1 << bt;
      for (int a = 0; a < 256; ++a) {
        if ((a & mc) && !(a & mt)) {
          int a2 = a | mt;
          float tr = sr[a]; sr[a] = sr[a2]; sr[a2] = tr;
          float ti = si[a]; si[a] = si[a2]; si[a2] = ti;
        }
      }
    }
  }
  for (int a = 0; a < 256; ++a) {
    float vr = sr[a], vi = si[a];
    float vv[4] = {vr, -vi, vi, vr};
    int rows[4] = {2 * a, 2 * a, 2 * a + 1, 2 * a + 1};
    int cols[4] = {2 * col, 2 * col + 1, 2 * col, 2 * col + 1};
    for (int e = 0; e < 4; ++e) {
      int mrow = rows[e], kk = cols[e];
      int mt2 = mrow >> 4, lr2 = mrow & 15;
      int ks = kk >> 2, rem = kk & 3;
      int lane2 = ((rem >> 1) << 4) + lr2, j2 = rem & 1;
      A4[((((size_t)g * 32 + mt2) * 128 + ks) * 32 + lane2) * 2 + j2] = vv[e];
    }
  }
}

// ---------------------------------------------------------------------------
// k1 (first step only): angles = [x_0, h0] @ W.T + b, then rank-1 product
// state psi0 with (-i)^popcount phase, stored in packed B-operand layout.
// ---------------------------------------------------------------------------
__global__ __launch_bounds__(64) void qlstm_angles_psi(
    const float* __restrict__ xall, int t, const float* __restrict__ H,
    QParams P, float* __restrict__ X) {
  int sample = blockIdx.x & (BATCH - 1);
  int g = blockIdx.x >> 7;
  __shared__ float red[NQ][8];
  __shared__ float ch[NQ], sh[NQ];
  int qq = threadIdx.x >> 3, pp = threadIdx.x & 7;
  {
    const float* wrow = P.W[g] + qq * COMB;
    const float* xrow = xall + ((size_t)t * BATCH + sample) * IN_DIM;
    const float* hrow = H + (size_t)sample * HID;
    float acc = 0.f;
    for (int k = pp * 48; k < pp * 48 + 48; ++k)
      acc += ((k < IN_DIM) ? xrow[k] : hrow[k - IN_DIM]) * wrow[k];
    red[qq][pp] = acc;
  }
  __syncthreads();
  if (threadIdx.x < NQ) {
    float s = P.b[g][threadIdx.x];
    for (int p2 = 0; p2 < 8; ++p2) s += red[threadIdx.x][p2];
    float ha = 0.5f * s;
    ch[threadIdx.x] = cosf(ha);
    sh[threadIdx.x] = sinf(ha);
  }
  __syncthreads();
  v2f* PB = (v2f*)X;
  int lr = sample & 15, nt = sample >> 4;
  for (int a = threadIdx.x; a < DIM; a += 64) {
    float mag = 1.f;
    int pc = 0;
    #pragma unroll
    for (int q2 = 0; q2 < NQ; ++q2) {
      int bit = (a >> (7 - q2)) & 1;
      mag *= bit ? sh[q2] : ch[q2];
      pc += bit;
    }
    int pm = pc & 3;
    float re = (pm == 0) ? mag : ((pm == 2) ? -mag : 0.f);
    float im = (pm == 1) ? -mag : ((pm == 3) ? mag : 0.f);
    store_psi(PB, g, nt, lr, a, re, im);
  }
}

// ---------------------------------------------------------------------------
// k2: WMMA GEMM. One wave per 16x16 output tile; 1024 tiles (4 gates x 32
// m-tiles x 8 n-tiles); K=512 -> 128 v_wmma_f32_16x16x4_f32 per wave, with
// one b64 A load + one b64 B load per WMMA. Expectation partials computed
// from the accumulator layout, reduced via shfl, written per m-tile.
// ---------------------------------------------------------------------------
__global__ __launch_bounds__(128) void qlstm_gemm_expect(
    const float* __restrict__ A4, const float* __restrict__ X,
    float* __restrict__ EP) {
  int wave = threadIdx.x >> 5;
  int lane = threadIdx.x & 31;
  int tile = blockIdx.x * 4 + wave;  // 0..1023
  int g  = tile >> 8;
  int tt = tile & 255;
  int mt = tt >> 3;
  int nt = tt & 7;
  int lr = lane & 15;
  int n  = nt * 16 + lr;
  const v2f* __restrict__ Ap =
      (const v2f*)A4 + ((size_t)(g * 32 + mt) * 128) * 32 + lane;
  const v2f* __restrict__ Bp =
      (const v2f*)X + ((size_t)(g * 8 + nt) * 128) * 32 + lane;
  v8f acc = {0.f, 0.f, 0.f, 0.f, 0.f, 0.f, 0.f, 0.f};
  #pragma unroll 8
  for (int ks = 0; ks < 128; ++ks) {
    v2f a = Ap[(size_t)ks * 32];
    v2f b = Bp[(size_t)ks * 32];
    acc = __builtin_amdgcn_wmma_f32_16x16x4_f32(false, a, false, b,
                                                (short)0, acc, false, false);
  }
  // C/D layout: vgpr v, lane l -> row mt*16 + v + 8*(l>>4), col n
  int rbase = mt * 16 + ((lane >> 4) << 3);
  float part[NQ];
  #pragma unroll
  for (int q2 = 0; q2 < NQ; ++q2) part[q2] = 0.f;
  #pragma unroll
  for (int j = 0; j < 4; ++j) {
    float re = acc[2 * j];
    float im = acc[2 * j + 1];
    float p = re * re + im * im;
    int amp = (rbase >> 1) + j;
    #pragma unroll
    for (int q2 = 0; q2 < NQ; ++q2)
      part[q2] += ((amp >> (7 - q2)) & 1) ? -p : p;
  }
  #pragma unroll
  for (int q2 = 0; q2 < NQ; ++q2) {
    float v = part[q2];
    v += __shfl_down(v, 16, 32);
    if (lane < 16)
      EP[((size_t)(g * 32 + mt) * NQ + q2) * 128 + n] = v;
  }
}

// ---------------------------------------------------------------------------
// k3 (fused): reduce expectation partials (deterministic), project 8->256
// through U, LSTM nonlinearities, update h/c, write outputs — then, with the
// fresh h row already in LDS, compute next step's angles and packed psi0.
// ---------------------------------------------------------------------------
__global__ __launch_bounds__(256) void qlstm_cell_fuse(
    const float* __restrict__ EP, QParams P,
    float* __restrict__ H, float* __restrict__ C,
    float* __restrict__ out, int t,
    const float* __restrict__ xall, float* __restrict__ X) {
  int sample = blockIdx.x;
  int j = threadIdx.x;
  __shared__ float es[4][NQ];
  __shared__ float hs[HID];
  __shared__ float red[4][NQ][8];
  __shared__ float ch[4][NQ], sh[4][NQ];
  if (j < 32) {
    int g = j >> 3, qq = j & 7;
    float s = 0.f;
    for (int mt = 0; mt < 32; ++mt)
      s += EP[((size_t)(g * 32 + mt) * NQ + qq) * 128 + sample];
    es[g][qq] = s;
  }
  __syncthreads();
  float vals[4];
  #pragma unroll
  for (int g = 0; g < 4; ++g) {
    const float* U = P.U[g];
    float acc = P.c[g][j];
    #pragma unroll
    for (int qq = 0; qq < NQ; ++qq) acc += es[g][qq] * U[j * NQ + qq];
    vals[g] = acc;
  }
  float f  = 1.f / (1.f + expf(-vals[0]));
  float i_ = 1.f / (1.f + expf(-vals[1]));
  float gg = tanhf(vals[2]);
  float o  = 1.f / (1.f + expf(-vals[3]));
  int idx = sample * HID + j;
  float cn = f * C[idx] + i_ * gg;
  float hn = o * tanhf(cn);
  C[idx] = cn;
  H[idx] = hn;
  hs[j] = hn;
  out[((size_t)t * BATCH + sample) * HID + j] = hn;
  if (t == SEQ - 1) {
    out[(size_t)SEQ * BATCH * HID + idx] = hn;                        // hx
    out[(size_t)SEQ * BATCH * HID + (size_t)BATCH * HID + idx] = cn;  // cx
    return;  // uniform across block
  }
  __syncthreads();
  // ---- next step's angles: 4 gates x 8 qubits, 8 threads x 48 elems each
  {
    int g2 = j >> 6, qq = (j >> 3) & 7, pp = j & 7;
    const float* wrow = P.W[g2] + qq * COMB;
    const float* xrow = xall + ((size_t)(t + 1) * BATCH + sample) * IN_DIM;
    float acc = 0.f;
    for (int k = pp * 48; k < pp * 48 + 48; ++k)
      acc += ((k < IN_DIM) ? xrow[k] : hs[k - IN_DIM]) * wrow[k];
    red[g2][qq][pp] = acc;
  }
  __syncthreads();
  if (j < 32) {
    int g3 = j >> 3, q3 = j & 7;
    float s = P.b[g3][q3];
    for (int p2 = 0; p2 < 8; ++p2) s += red[g3][q3][p2];
    float ha = 0.5f * s;
    ch[g3][q3] = cosf(ha);
    sh[g3][q3] = sinf(ha);
  }
  __syncthreads();
  // ---- next step's packed psi0
  v2f* PB = (v2f*)X;
  int lr = sample & 15, nt = sample >> 4;
  for (int id2 = j; id2 < 4 * DIM; id2 += 256) {
    int g3 = id2 >> 8, a = id2 & 255;
    float mag = 1.f;
    int pc = 0;
    #pragma unroll
    for (int q2 = 0; q2 < NQ; ++q2) {
      int bit = (a >> (7 - q2)) & 1;
      mag *= bit ? sh[g3][q2] : ch[g3][q2];
      pc += bit;
    }
    int pm = pc & 3;
    float re = (pm == 0) ? mag : ((pm == 2) ? -mag : 0.f);
    float im = (pm == 1) ? -mag : ((pm == 3) ? mag : 0.f);
    store_psi(PB, g3, nt, lr, a, re, im);
  }
}

extern "C" void kernel_launch(void* const* d_in, const int* in_sizes, int n_in,
                              void* d_out, int out_size, void* d_ws, size_t ws_size,
                              hipStream_t stream) {
  (void)in_sizes; (void)n_in; (void)out_size; (void)ws_size;
  const float* inputs = (const float*)d_in[0];
  QParams P;
  for (int g = 0; g < 4; ++g) {
    P.W[g] = (const float*)d_in[1 + 5 * g];
    P.b[g] = (const float*)d_in[2 + 5 * g];
    P.q[g] = (const float*)d_in[3 + 5 * g];
    P.U[g] = (const float*)d_in[4 + 5 * g];
    P.c[g] = (const float*)d_in[5 + 5 * g];
  }
  float* ws = (float*)d_ws;
  float* A4 = ws + OFF_A4;
  float* X  = ws + OFF_X;
  float* EP = ws + OFF_EP;
  float* H  = ws + OFF_H;
  float* C  = ws + OFF_C;
  float* out = (float*)d_out;

  qlstm_zero<<<256, 256, 0, stream>>>(H, 2 * BATCH * HID);  // H,C contiguous
  qlstm_build_unitary<<<16, 64, 0, stream>>>(P, A4);
  qlstm_angles_psi<<<4 * BATCH, 64, 0, stream>>>(inputs, 0, H, P, X);
  for (int t = 0; t < SEQ; ++t) {
    qlstm_gemm_expect<<<256, 128, 0, stream>>>(A4, X, EP);
    qlstm_cell_fuse<<<BATCH, 256, 0, stream>>>(EP, P, H, C, out, t, inputs, X);
  }
}